// Netwoc1_86517821210770
// MI455X (gfx1250) — compile-verified
//
#include <hip/hip_runtime.h>
#include <math.h>

typedef float v2f __attribute__((ext_vector_type(2)));
typedef float v4f __attribute__((ext_vector_type(4)));
typedef float v8f __attribute__((ext_vector_type(8)));

typedef int v4i __attribute__((vector_size(4 * sizeof(int))));
typedef __attribute__((address_space(1))) v4i* gp128;   // global int4*
typedef __attribute__((address_space(3))) v4i* lp128;   // LDS int4*

#if defined(__has_builtin)
#  if __has_builtin(__builtin_amdgcn_global_load_async_to_lds_b128) && \
      __has_builtin(__builtin_amdgcn_s_wait_asynccnt)
#    define USE_ASYNC_LDS 1
#  endif
#endif
#ifndef USE_ASYNC_LDS
#  define USE_ASYNC_LDS 0
#endif

#define N_BATCH 1024
#define C_IN    80
#define T_LEN   500
#define CP1     81
#define TCHUNK  64
#define NCHUNKS 8            // ceil(500/64); last chunk has 52 valid t
#define TSTR    72           // [r][t] row stride: 16B-aligned, 2*TSTR==16 (mod 64) banks
#define RROWS   160          // padded r rows: r = c+32, halo zeros
#define XBUF    (RROWS * TSTR)
#define YSTR    65           // odd -> conflict-free scan reads
#define MROWS   96
#define YBUF    (MROWS * YSTR)

__global__ __launch_bounds__(512) void snn_fused_kernel(
    const float* __restrict__ x,
    const float* __restrict__ conv_w,
    const float* __restrict__ conv_b,
    const float* __restrict__ bn_gamma,
    const float* __restrict__ bn_beta,
    const float* __restrict__ bn_mean,
    const float* __restrict__ bn_var,
    const float* __restrict__ plif_w,
    const float* __restrict__ fc_w,
    const float* __restrict__ fc_b,
    float* __restrict__ out)
{
    __shared__ float xpad_s[2 * XBUF];   // 92,160 B : double-buffered x tile, [r][t]
    __shared__ float y_s[2 * YBUF];      // 49,920 B : double-buffered conv+BN tile, [c'][t]
    __shared__ float wpad_s[192];        // zero-padded taps: [64..128) = w[0..64)
    __shared__ float inv_s[MROWS];
    __shared__ float add_s[MROWS];
    __shared__ float feat_s[CP1];

    const int tid  = threadIdx.x;
    const int n    = blockIdx.x;
    const int lane = tid & 31;
    const int wave = tid >> 5;

    // ---- per-block setup ----
    for (int i = tid; i < 192; i += 512)
        wpad_s[i] = (i >= 64 && i < 128) ? conv_w[i - 64] : 0.0f;
    for (int i = tid; i < MROWS; i += 512) {
        if (i < CP1) {
            float inv = bn_gamma[i] * rsqrtf(bn_var[i] + 1e-5f);
            inv_s[i] = inv;
            add_s[i] = (conv_b[0] - bn_mean[i]) * inv + bn_beta[i];
        } else { inv_s[i] = 0.0f; add_s[i] = 0.0f; }
    }
    // zero both x buffers once; halo rows stay zero, interior rewritten per chunk
    for (int i = tid; i < 2 * XBUF; i += 512) xpad_s[i] = 0.0f;

    const float decay = 1.0f / (1.0f + __expf(-plif_w[0]));
    const float* gsrc = x + (size_t)n * (C_IN * T_LEN);

    // lane-constant WMMA fragment addressing (wave32, 16x16x4 f32 layouts)
    const int m_l   = lane & 15;          // A row / B,C column within tile
    const int kkb   = (lane >> 4) << 1;   // K sub-offset: 0 (lanes 0-15) or 2 (lanes 16-31)
    const int abase = 64 + kkb - m_l;     // wpad idx: Wmat[m0+m, m0+4s+kk] = w[4s+kk-m]

    float v_state = 0.0f;   // PLIF membrane (threads 0..80)
    float cnt     = 0.0f;   // spike count

    __syncthreads();

    // ======== 3-stage pipeline: load chunk i | conv chunk i-1 | scan chunk i-2 ========
    for (int i = 0; i < NCHUNKS + 2; ++i) {

        // ---- stage A: issue async global->LDS loads for chunk i ----
        if (i < NCHUNKS) {
            float* xb = &xpad_s[(i & 1) * XBUF];
            const int t0 = i * TCHUNK;
            for (int e = tid; e < C_IN * (TCHUNK / 4); e += 512) {   // 1280 x b128
                int c  = e >> 4;
                int tl = (e & 15) << 2;
                int t  = t0 + tl;
                float* ldst = &xb[(c + 32) * TSTR + tl];
                if (t + 3 < T_LEN) {                 // T=500: groups are all-valid or all-invalid
#if USE_ASYNC_LDS
                    __builtin_amdgcn_global_load_async_to_lds_b128(
                        (gp128)(gsrc + c * T_LEN + t), (lp128)ldst, 0, 0);
#else
                    *(v4f*)ldst = *(const v4f*)(gsrc + c * T_LEN + t);
#endif
                } else {
                    v4f z = {0.0f, 0.0f, 0.0f, 0.0f};
                    *(v4f*)ldst = z;
                }
            }
        }

        // ---- stage B: banded GEMM via f32 WMMA on chunk i-1 ----
        // Y[96 x 64] = Wmat[96 x K] * Xpad[K x 64]; 6 m-tiles x 4 t-tiles = 24 jobs
        if (i >= 1 && i <= NCHUNKS) {
            const float* xb = &xpad_s[((i - 1) & 1) * XBUF];
            float*       yb = &y_s[((i - 1) & 1) * YBUF];
            for (int job = wave; job < 24; job += 16) {
                const int m0  = (job >> 2) << 4;   // 0,16,...,80
                const int tl0 = (job & 3) << 4;    // 0,16,32,48
                v8f acc = {};
                #pragma unroll 4
                for (int s = 0; s < 20; ++s) {     // K = 80 band, steps of 4
                    const int r = m0 + 4 * s + kkb;
                    v2f a, b;
                    a.x = wpad_s[abase + 4 * s];
                    a.y = wpad_s[abase + 4 * s + 1];
                    b.x = xb[r * TSTR + tl0 + m_l];
                    b.y = xb[(r + 1) * TSTR + tl0 + m_l];
                    acc = __builtin_amdgcn_wmma_f32_16x16x4_f32(
                            false, a, false, b, (short)0, acc, false, false);
                }
                #pragma unroll
                for (int j = 0; j < 8; ++j) {      // fused BatchNorm epilogue
                    int row = m0 + j + ((lane >> 4) << 3);
                    yb[row * YSTR + tl0 + m_l] = acc[j] * inv_s[row] + add_s[row];
                }
            }
        }

        // ---- stage C: PLIF scan on chunk i-2 (81 independent chains) ----
        if (i >= 2 && tid < CP1) {
            const int cidx = i - 2;
            int tvalid = T_LEN - cidx * TCHUNK;
            if (tvalid > TCHUNK) tvalid = TCHUNK;
            const float* yr = &y_s[(cidx & 1) * YBUF + tid * YSTR];
            float v = v_state, cn = cnt;
            for (int tl = 0; tl < tvalid; ++tl) {
                float xt = yr[tl];
                v += (xt - v) * decay;
                float s = (v >= 1.0f) ? 1.0f : 0.0f;
                cn += s;
                v *= (1.0f - s);
            }
            v_state = v; cnt = cn;
        }

#if USE_ASYNC_LDS
        __builtin_amdgcn_s_wait_asynccnt(0);
#endif
        __syncthreads();
    }

    // ---- mean over T, then 81->3 FC ----
    if (tid < CP1) feat_s[tid] = cnt * (1.0f / (float)T_LEN);
    __syncthreads();
    if (tid < 3) {
        float o = fc_b[tid];
        for (int c = 0; c < CP1; ++c)
            o += feat_s[c] * fc_w[tid * CP1 + c];
        out[n * 3 + tid] = o;
    }
}

extern "C" void kernel_launch(void* const* d_in, const int* in_sizes, int n_in,
                              void* d_out, int out_size, void* d_ws, size_t ws_size,
                              hipStream_t stream) {
    (void)in_sizes; (void)n_in; (void)out_size; (void)d_ws; (void)ws_size;
    const float* x        = (const float*)d_in[0];
    const float* conv_w   = (const float*)d_in[1];
    const float* conv_b   = (const float*)d_in[2];
    const float* bn_gamma = (const float*)d_in[3];
    const float* bn_beta  = (const float*)d_in[4];
    const float* bn_mean  = (const float*)d_in[5];
    const float* bn_var   = (const float*)d_in[6];
    const float* plif_w   = (const float*)d_in[7];
    const float* fc_w     = (const float*)d_in[8];
    const float* fc_b     = (const float*)d_in[9];
    float* out = (float*)d_out;

    snn_fused_kernel<<<dim3(N_BATCH), dim3(512), 0, stream>>>(
        x, conv_w, conv_b, bn_gamma, bn_beta, bn_mean, bn_var,
        plif_w, fc_w, fc_b, out);
}